// BetterParaformerV2_3521873183300
// MI455X (gfx1250) — compile-verified
//
#include <hip/hip_runtime.h>

#define B_ 16
#define T_ 1024
#define V_ 2048
#define FEAT_ 4100            // 2*V + 4
#define NBLANK 0

typedef __attribute__((ext_vector_type(2))) float v2f;
typedef __attribute__((ext_vector_type(8))) float v8f;

// d_out layout (floats): padded [B*T*FEAT] | piece_lengths [B] | piece_conf [B*T]
#define LEN_OFF  ((size_t)B_ * T_ * FEAT_)
#define CONF_OFF (LEN_OFF + B_)

// branch-proof gated select: gate bits AND'ed with compare mask (no predicated loads)
__device__ __forceinline__ float gsel(int2 m, int tgt) {
    return __int_as_float(m.y & -(int)(m.x == tgt));
}

// ---------------------------------------------------------------------------
// Phase 1: run-length encode, per-frame gates, per-rank descriptors, tail
// features, conf, piece_lengths.  One block (1024 threads) per batch item.
// ---------------------------------------------------------------------------
__global__ __launch_bounds__(1024) void seg_meta_kernel(
    const float* __restrict__ shallow, const float* __restrict__ finalp,
    const float* __restrict__ boundary, const int* __restrict__ align,
    const int* __restrict__ lengths, float* __restrict__ out,
    int* __restrict__ slot, float* __restrict__ gateArr,
    int* __restrict__ rank_start, int* __restrict__ rank_end,
    float* __restrict__ row_scale, int* __restrict__ nrank_arr)
{
    __shared__ int   s_lab[T_];
    __shared__ int   s_scan[T_];
    __shared__ float s_gsum[T_];
    __shared__ float s_csum[T_];
    __shared__ float s_ssum[T_];
    __shared__ float s_cnt[T_];
    __shared__ int   s_start[T_];
    __shared__ int   s_end[T_];
    __shared__ int   s_rank[T_];
    __shared__ int   s_nseg;

    const int b = blockIdx.x;
    const int t = threadIdx.x;
    const int len = lengths[b];
    const bool valid = (t < len);
    const int lab = valid ? align[b * T_ + t] : NBLANK;

    s_lab[t] = lab;
    s_gsum[t] = 0.f; s_csum[t] = 0.f; s_ssum[t] = 0.f; s_cnt[t] = 0.f;
    s_start[t] = T_; s_end[t] = -1;
    __syncthreads();

    // run-id = inclusive-scan(change) - 1
    int change = (t == 0) ? 1 : (lab != s_lab[t - 1] ? 1 : 0);
    s_scan[t] = change;
    __syncthreads();
    for (int off = 1; off < T_; off <<= 1) {
        int add = (t >= off) ? s_scan[t - off] : 0;
        __syncthreads();
        s_scan[t] += add;
        __syncthreads();
    }
    const int run = s_scan[t] - 1;

    // per-frame gate + run aggregates
    float gate = 0.f;
    if (valid) {
        size_t base = ((size_t)(b * T_ + t)) * V_;
        float slv = shallow[base + lab];
        float flv = finalp[base + lab];
        float fc  = 0.5f * (slv + flv);
        gate = 1.f / (1.f + __expf(-(fc - 0.55f) * 12.f)) + 0.05f;
        atomicAdd(&s_gsum[run], gate);
        atomicAdd(&s_csum[run], fc);
        atomicAdd(&s_ssum[run], slv);
        atomicAdd(&s_cnt[run], 1.f);
        atomicMin(&s_start[run], t);
        atomicMax(&s_end[run], t);
    }
    gateArr[b * T_ + t] = gate;
    __syncthreads();

    // run-level: thread index doubles as run index r
    const int   r      = t;
    const float cnt    = s_cnt[r];
    const int   startv = s_start[r];
    const int   labr   = (startv < T_) ? s_lab[startv] : NBLANK;
    const int   vr     = (cnt > 0.f && labr != NBLANK) ? 1 : 0;

    s_scan[t] = vr;                     // rank = inclusive-scan(valid_run) - 1
    __syncthreads();
    for (int off = 1; off < T_; off <<= 1) {
        int add = (t >= off) ? s_scan[t - off] : 0;
        __syncthreads();
        s_scan[t] += add;
        __syncthreads();
    }
    const int rank = s_scan[t] - 1;
    if (t == T_ - 1) s_nseg = s_scan[T_ - 1];
    s_rank[r] = vr ? rank : -1;
    __syncthreads();
    const int nseg = s_nseg;

    // per-frame output rank slot (-1 for blank / pad frames)
    slot[b * T_ + t] = valid ? s_rank[run] : -1;

    if (vr) {
        const int rend = s_end[r] + 1;
        const float denom    = fmaxf(cnt, 1.f);
        const float seg_conf = s_csum[r] / denom;
        const float smean    = s_ssum[r] / denom;
        const float left_b   = (rank == 0) ? 1.f
                             : boundary[b * T_ + max(startv - 1, 0)];
        const float right_b  = boundary[b * T_ + (rend - 1)];
        const float scale    = 1.f + 0.25f * (1.f - left_b) + 0.25f * (1.f - right_b);
        const float gs       = fmaxf(s_gsum[r], 1e-6f);
        const int ri = b * T_ + rank;
        row_scale[ri]  = scale / gs;
        rank_start[ri] = startv;
        rank_end[ri]   = rend;
        size_t ob = ((size_t)ri) * FEAT_ + 2 * V_;
        out[ob + 0] = seg_conf * scale;
        out[ob + 1] = smean * scale;
        out[ob + 2] = left_b * scale;
        out[ob + 3] = right_b * scale;
        out[CONF_OFF + ri] = fminf(fmaxf(seg_conf, 0.05f), 1.f);
    }
    if (t >= nseg) {                    // default-fill unused rank slots
        const int ri = b * T_ + t;
        row_scale[ri]  = 0.f;
        rank_start[ri] = T_;
        rank_end[ri]   = T_;
        size_t ob = ((size_t)ri) * FEAT_ + 2 * V_;
        out[ob + 0] = 0.f; out[ob + 1] = 0.f; out[ob + 2] = 0.f; out[ob + 3] = 0.f;
        out[CONF_OFF + ri] = (nseg == 0 && t == 0) ? 1.f : 0.f;
    }
    if (t == 0) {
        nrank_arr[b] = nseg;
        out[LEN_OFF + b] = (float)max(nseg, 1);
    }
}

// ---------------------------------------------------------------------------
// Phase 2: gate-weighted segment pooling as tiled WMMA.
//   D(16 ranks x 16 cols) += A(16 ranks x 4 frames) * B(4 frames x 16 cols)
// A is built from LDS-staged (slot,gate) int2 pairs; the gated select is a
// bitwise AND with the compare mask so no load can be predicated.  Main loop
// runs full 8-frame chunks with constant-offset B loads; one clamped tail
// trip handles the rest (LDS padding slot=-1 zeroes A there).  Rank tiles own
// contiguous frame ranges, so each output tile is written exactly once.
// ---------------------------------------------------------------------------
__global__ __launch_bounds__(256) void seg_pool_wmma_kernel(
    const float* __restrict__ shallow, const float* __restrict__ finalp,
    float* __restrict__ out,
    const int* __restrict__ slot, const float* __restrict__ gateArr,
    const int* __restrict__ rank_start, const int* __restrict__ rank_end,
    const float* __restrict__ row_scale, const int* __restrict__ nrank_arr)
{
    __shared__ float s_rs[16];
    __shared__ int2  s_meta[T_ + 8];   // .x = slot, .y = gate (float bits)

    const int b     = blockIdx.z;
    const int rbase = blockIdx.y * 16;
    const int wave  = threadIdx.x >> 5;
    const int lane  = threadIdx.x & 31;
    const int colTile = blockIdx.x * 8 + wave;   // 0..255 (128 shallow + 128 final)
    const int nrank = nrank_arr[b];

    const int n    = lane & 15;       // column within tile == A row m (lane layout)
    const int half = lane >> 4;       // K-half / M-half selector
    const int outcol = colTile * 16;  // shallow: 0..2047, final: 2048..4095

    if (rbase >= nrank) {             // block-uniform: tile past last segment
        #pragma unroll
        for (int v = 0; v < 8; ++v) {
            int row = rbase + half * 8 + v;
            out[((size_t)(b * T_ + row)) * FEAT_ + outcol + n] = 0.f;
        }
        return;
    }

    const int fstart = rank_start[b * T_ + rbase];
    const int lastr  = min(rbase + 15, nrank - 1);
    const int fend   = rank_end[b * T_ + lastr];
    const int range  = fend - fstart;            // block-uniform, <= T_

    // stage per-frame metadata for this rank tile's frame range into LDS
    for (int i = threadIdx.x; i < range; i += 256) {
        s_meta[i] = make_int2(slot[b * T_ + fstart + i],
                              __float_as_int(gateArr[b * T_ + fstart + i]));
    }
    if (threadIdx.x < 8)                          // tail padding: kills OOB frames
        s_meta[range + threadIdx.x] = make_int2(-1, 0);
    if (threadIdx.x < 16)
        s_rs[threadIdx.x] = row_scale[b * T_ + rbase + threadIdx.x];
    __syncthreads();

    const float* src = (colTile < 128) ? shallow : finalp;
    const int k0  = half * 2;
    const int tgt = rbase + n;
    // per-lane source pointer: row fstart+k0, column (colTile&127)*16 + n
    const float* q = src + (size_t)(b * T_ + fstart + k0) * V_
                         + (size_t)((colTile & 127) * 16 + n);

    v8f c = {};
    int f0 = 0;
    for (; f0 + 8 <= range; f0 += 8, q += 8 * V_) {   // full chunks: no clamping
        const int2 m0 = s_meta[f0 + k0];
        const int2 m1 = s_meta[f0 + k0 + 1];
        const int2 m2 = s_meta[f0 + 4 + k0];
        const int2 m3 = s_meta[f0 + 4 + k0 + 1];
        v2f a0, a1, b0, b1;
        a0.x = gsel(m0, tgt);
        a0.y = gsel(m1, tgt);
        a1.x = gsel(m2, tgt);
        a1.y = gsel(m3, tgt);
        b0.x = q[0];
        b0.y = q[V_];
        b1.x = q[4 * V_];
        b1.y = q[5 * V_];
        c = __builtin_amdgcn_wmma_f32_16x16x4_f32(
                false, a0, false, b0, (short)0, c, false, false);
        c = __builtin_amdgcn_wmma_f32_16x16x4_f32(
                false, a1, false, b1, (short)0, c, false, false);
    }
    if (f0 < range) {                                 // clamped tail trip
        const int2 m0 = s_meta[f0 + k0];
        const int2 m1 = s_meta[f0 + k0 + 1];
        const int2 m2 = s_meta[f0 + 4 + k0];
        const int2 m3 = s_meta[f0 + 4 + k0 + 1];
        const size_t base = (size_t)b * T_ * V_ + (size_t)((colTile & 127) * 16 + n);
        v2f a0, a1, b0, b1;
        a0.x = gsel(m0, tgt);
        a0.y = gsel(m1, tgt);
        a1.x = gsel(m2, tgt);
        a1.y = gsel(m3, tgt);
        b0.x = src[base + (size_t)min(fstart + f0 + k0,     T_ - 1) * V_];
        b0.y = src[base + (size_t)min(fstart + f0 + k0 + 1, T_ - 1) * V_];
        b1.x = src[base + (size_t)min(fstart + f0 + 4 + k0,     T_ - 1) * V_];
        b1.y = src[base + (size_t)min(fstart + f0 + 4 + k0 + 1, T_ - 1) * V_];
        c = __builtin_amdgcn_wmma_f32_16x16x4_f32(
                false, a0, false, b0, (short)0, c, false, false);
        c = __builtin_amdgcn_wmma_f32_16x16x4_f32(
                false, a1, false, b1, (short)0, c, false, false);
    }

    #pragma unroll
    for (int v = 0; v < 8; ++v) {
        int row = rbase + half * 8 + v;                 // D layout: M = v + 8*half
        float val = c[v] * s_rs[half * 8 + v];          // fold scale/gate_sum per row
        out[((size_t)(b * T_ + row)) * FEAT_ + outcol + n] = val;
    }
}

// ---------------------------------------------------------------------------
extern "C" void kernel_launch(void* const* d_in, const int* in_sizes, int n_in,
                              void* d_out, int out_size, void* d_ws, size_t ws_size,
                              hipStream_t stream) {
    (void)in_sizes; (void)n_in; (void)out_size; (void)ws_size;
    const float* shallow  = (const float*)d_in[0];
    const float* finalp   = (const float*)d_in[1];
    const float* boundary = (const float*)d_in[2];
    const int*   align    = (const int*)d_in[3];
    const int*   lengths  = (const int*)d_in[4];
    float* out = (float*)d_out;

    char* ws = (char*)d_ws;
    const size_t N = (size_t)B_ * T_;
    int*   slot       = (int*)  (ws);
    float* gateArr    = (float*)(ws + 1 * N * 4);
    int*   rank_start = (int*)  (ws + 2 * N * 4);
    int*   rank_end   = (int*)  (ws + 3 * N * 4);
    float* row_scale  = (float*)(ws + 4 * N * 4);
    int*   nrank_arr  = (int*)  (ws + 5 * N * 4);

    seg_meta_kernel<<<B_, 1024, 0, stream>>>(
        shallow, finalp, boundary, align, lengths, out,
        slot, gateArr, rank_start, rank_end, row_scale, nrank_arr);

    dim3 grid(32, T_ / 16, B_);   // 32 col-tile groups x 64 rank tiles x 16 batches
    seg_pool_wmma_kernel<<<grid, 256, 0, stream>>>(
        shallow, finalp, out,
        slot, gateArr, rank_start, rank_end, row_scale, nrank_arr);
}